// Wav2TensorCore_10187662426699
// MI455X (gfx1250) — compile-verified
//
#include <hip/hip_runtime.h>
#include <hip/hip_bf16.h>
#include <math.h>

// ---------------- problem constants ----------------
#define B_      8
#define T_IN    661500
#define NFFT    1024
#define HOP_    256
#define F_      513          // NFFT/2 + 1
#define NT      2584         // 1 + (T_IN)/HOP
#define FP      528          // F_ padded to 33*16
#define NTILE_M 33           // FP/16
#define NTILE_N 162          // ceil(NT/16)
#define KPAD    516          // F_ padded to multiple of 4 (psy GEMM K)

typedef __attribute__((ext_vector_type(2))) float v2f;
typedef __attribute__((ext_vector_type(8))) float v8f;
typedef __attribute__((ext_vector_type(4))) unsigned int v4u;
typedef __attribute__((ext_vector_type(8))) unsigned int v8u;

// ---------------- fill DFT basis (window folded in) ----------------
__global__ __launch_bounds__(256) void k_fill_basis(float* __restrict__ bre,
                                                    float* __restrict__ bim) {
  int idx = blockIdx.x * blockDim.x + threadIdx.x;
  if (idx >= FP * NFFT) return;
  int f = idx >> 10;          // row
  int k = idx & (NFFT - 1);   // col
  float vre = 0.f, vim = 0.f;
  if (f < F_) {
    const float TWO_PI = 6.28318530717958647692f;
    float w = 0.5f * (1.0f - cosf(TWO_PI * (float)k / (float)NFFT));  // periodic hann
    int r = (f * k) & (NFFT - 1);                                     // exact angle mod 2pi
    float ang = TWO_PI * (float)r / (float)NFFT;
    vre =  w * cosf(ang);
    vim = -w * sinf(ang);
  }
  bre[idx] = vre;
  bim[idx] = vim;
}

// ---------------- fill bark spread matrix (row-normalized) ----------------
__device__ __forceinline__ float bark_of(int g) {
  float fr = (float)g * (11025.0f / 512.0f);   // linspace(0, SR/2, F_)
  float a = fr / 7500.0f;
  return 13.0f * atanf(0.00076f * fr) + 3.5f * atanf(a * a);
}

__global__ __launch_bounds__(256) void k_fill_spread(float* __restrict__ spread) {
  __shared__ float row[FP];
  __shared__ float red[256];
  int f = blockIdx.x;
  int tid = threadIdx.x;
  float bf = (f < F_) ? bark_of(f) : 0.f;
  float s = 0.f;
  for (int g = tid; g < FP; g += 256) {
    float m = 0.f;
    if (f < F_ && g < F_) {
      float d = fabsf(bf - bark_of(g));
      // d==0 -> exp(-0.5 d^2)=1 ; d>0 -> exp(-0.7*(d/0.5)^2)=exp(-2.8 d^2). Agree at 0.
      m = expf(-2.8f * d * d);
    }
    row[g] = m;
    s += m;
  }
  red[tid] = s;
  __syncthreads();
  for (int o = 128; o > 0; o >>= 1) {
    if (tid < o) red[tid] += red[tid + o];
    __syncthreads();
  }
  float inv = 1.0f / (red[0] + 1e-8f);
  for (int g = tid; g < FP; g += 256) spread[(size_t)f * FP + g] = row[g] * inv;
}

// ---------------- fused STFT via fp32 WMMA ----------------
// grid: (NTILE_N, B_), block 256 (8 waves). Each block: one 16-frame strip,
// both channels staged in LDS (reflect-pad gather, so staged manually);
// each wave walks M-tiles. Per K-step: 4 WMMAs (L/R x re/im).
__global__ __launch_bounds__(256) void k_stft(const float* __restrict__ x,
                                              const float* __restrict__ bre,
                                              const float* __restrict__ bim,
                                              float* __restrict__ out,
                                              float* __restrict__ mag,
                                              float* __restrict__ mdb) {
  extern __shared__ float lds[];                 // 2 * NFFT * 17 floats
  const int STRIP = NFFT * 17;
  int tid = threadIdx.x;
  int lane = tid & 31, wave = tid >> 5;
  int tbase = blockIdx.x * 16;
  int b = blockIdx.y;

  // stage both channels' frame strip: lds[ch][k][t0], row pad 17
  for (int idx = tid; idx < 2 * NFFT * 16; idx += 256) {
    int ch = idx >> 14;
    int rem = idx & 16383;
    int k = rem & (NFFT - 1);
    int t0 = rem >> 10;
    int t = tbase + t0;
    float v = 0.f;
    if (t < NT) {
      int j = t * HOP_ + k - (NFFT / 2);         // reflect pad
      if (j < 0) j = -j;
      if (j >= T_IN) j = 2 * (T_IN - 1) - j;
      v = x[((size_t)b * 2 + ch) * T_IN + j];
    }
    lds[ch * STRIP + k * 17 + t0] = v;
  }
  __syncthreads();

  int lane16 = lane & 15;
  int khi = (lane >> 4) << 1;                    // 0 or 2 (A/B K sub-lane split)

  for (int m = wave; m < NTILE_M; m += 8) {
    int fbase = m * 16;
    const float* arow_re = bre + (size_t)(fbase + lane16) * NFFT;
    const float* arow_im = bim + (size_t)(fbase + lane16) * NFFT;
    v8f cLre = {}, cLim = {}, cRre = {}, cRim = {};

    for (int kb = 0; kb < NFFT; kb += 4) {
      if ((kb & 63) == 0) {
        __builtin_prefetch(arow_re + kb + 64, 0, 1);   // global_prefetch_b8
        __builtin_prefetch(arow_im + kb + 64, 0, 1);
      }
      int k0 = kb + khi;
      v2f are = { arow_re[k0], arow_re[k0 + 1] };
      v2f aim = { arow_im[k0], arow_im[k0 + 1] };
      v2f bL  = { lds[k0 * 17 + lane16],          lds[(k0 + 1) * 17 + lane16] };
      v2f bR  = { lds[STRIP + k0 * 17 + lane16],  lds[STRIP + (k0 + 1) * 17 + lane16] };
      cLre = __builtin_amdgcn_wmma_f32_16x16x4_f32(false, are, false, bL, (short)0, cLre, false, false);
      cLim = __builtin_amdgcn_wmma_f32_16x16x4_f32(false, aim, false, bL, (short)0, cLim, false, false);
      cRre = __builtin_amdgcn_wmma_f32_16x16x4_f32(false, are, false, bR, (short)0, cRre, false, false);
      cRim = __builtin_amdgcn_wmma_f32_16x16x4_f32(false, aim, false, bR, (short)0, cRim, false, false);
    }

    // epilogue: C/D layout — VGPR r: lanes 0-15 -> M=r, lanes 16-31 -> M=r+8
    int t = tbase + lane16;
    int frow = fbase + ((lane >> 4) << 3);
    if (t < NT) {
      #pragma unroll
      for (int r = 0; r < 8; ++r) {
        int f = frow + r;
        if (f < F_) {
          float Lre = cLre[r], Lim = cLim[r], Rre = cRre[r], Rim = cRim[r];
          float sre = 0.5f * (Lre + Rre);
          float sim = 0.5f * (Lim + Rim);
          float mg = sqrtf(sre * sre + sim * sim);
          float md = fmaxf(mg, 1e-10f);
          float db = 20.0f * log10f(md);
          db = fminf(fmaxf(db, -100.0f), 0.0f);
          db = (db + 100.0f) * 0.01f;
          float ipd = atan2f(Lim, Lre) - atan2f(Rim, Rre);
          float el = Lre * Lre + Lim * Lim;
          float er = Rre * Rre + Rim * Rim;
          float pan = (el - er) / (el + er + 1e-10f);
          size_t pstride = (size_t)F_ * NT;
          size_t base0 = ((size_t)b * 6) * pstride + (size_t)f * NT + t;
          out[base0]               = sre;   // plane 0: real
          out[base0 + pstride]     = sim;   // plane 1: imag
          out[base0 + 3 * pstride] = ipd;   // plane 3: ipd
          out[base0 + 4 * pstride] = pan;   // plane 4: panning
          size_t mo = ((size_t)b * F_ + f) * NT + t;
          mag[mo] = mg;
          mdb[mo] = db;
        }
      }
    }
  }
}

// ---------------- psychoacoustic GEMM via fp32 WMMA + TDM staging ----------------
// B-operand strip (mag_db[g][tbase..tbase+15]) is a clean 2D tile:
// dim0 = t (contiguous, 16 elems), dim1 = g (528 rows, stride NT).
// TDM handles OOB zero-fill (t tail via tensor_dim0, g>=513 via tensor_dim1)
// and inserts 1 DWORD pad per 16 DWORDs -> 17-float LDS pitch (bank-conflict free).
__global__ __launch_bounds__(256) void k_psy(const float* __restrict__ spread,
                                             const float* __restrict__ mdb,
                                             float* __restrict__ out) {
  __shared__ float lds_s[FP * 17];               // sole LDS object -> offset 0
  int tid = threadIdx.x;
  int lane = tid & 31, wave = tid >> 5;
  int tbase = blockIdx.x * 16;
  int b = blockIdx.y;

  if (wave == 0) {
    unsigned long long ga =
        (unsigned long long)(const void*)(mdb + ((size_t)b * F_) * NT + tbase);
    unsigned int td0 = (unsigned int)(NT - tbase);   // valid t from tile start
    v4u g0;
    g0[0] = 1u;                                      // count=1 (valid descriptor)
    g0[1] = 0u;                                      // lds_addr = 0
    g0[2] = (unsigned int)(ga & 0xFFFFFFFFu);        // global_addr[31:0]
    g0[3] = (unsigned int)((ga >> 32) & 0x01FFFFFFu) | (2u << 30);  // addr hi | type=2
    v8u g1;
    g1[0] = (2u << 16) | (1u << 20) | (3u << 22);    // data_size=4B, pad_en, ivl=16DW, amt=1DW
    g1[1] = (td0 & 0xFFFFu) << 16;                   // tensor_dim0[15:0] @ bit48
    g1[2] = (td0 >> 16) | ((unsigned int)F_ << 16);  // tensor_dim0 hi | tensor_dim1 lo
    g1[3] = (16u << 16);                             // tensor_dim1 hi(0) | tile_dim0=16
    g1[4] = (unsigned int)FP;                        // tile_dim1=528, tile_dim2=0
    g1[5] = (unsigned int)NT;                        // tensor_dim0_stride lo
    g1[6] = 0u;                                      // stride hi | dim1_stride lo
    g1[7] = 0u;
    asm volatile("tensor_load_to_lds %0, %1" :: "s"(g0), "s"(g1) : "memory");
    __builtin_amdgcn_s_wait_tensorcnt(0);
  }
  __syncthreads();

  int lane16 = lane & 15;
  int khi = (lane >> 4) << 1;

  for (int m = wave; m < NTILE_M; m += 8) {
    int fbase = m * 16;
    const float* arow = spread + (size_t)(fbase + lane16) * FP;
    v8f acc = {};
    for (int kb = 0; kb < KPAD; kb += 4) {
      int k0 = kb + khi;
      v2f a  = { arow[k0], arow[k0 + 1] };
      v2f bv = { lds_s[k0 * 17 + lane16], lds_s[(k0 + 1) * 17 + lane16] };
      acc = __builtin_amdgcn_wmma_f32_16x16x4_f32(false, a, false, bv, (short)0, acc, false, false);
    }
    int t = tbase + lane16;
    int frow = fbase + ((lane >> 4) << 3);
    if (t < NT) {
      #pragma unroll
      for (int r = 0; r < 8; ++r) {
        int f = frow + r;
        if (f < F_) {
          float v = acc[r] - 0.12f;
          v = fminf(fmaxf(v, 0.f), 1.f);
          out[(((size_t)b * 6 + 5) * F_ + f) * NT + t] = v;   // plane 5
        }
      }
    }
  }
}

// ---------------- harmonic plane (analytic down/up interpolation) ----------------
__device__ __forceinline__ float down_val(const float* __restrict__ magbt, int j, int Fk) {
  float sd = ((float)j + 0.5f) * ((float)F_ / (float)Fk) - 0.5f;
  sd = fminf(fmaxf(sd, 0.f), (float)(F_ - 1));
  int i0 = (int)sd;
  float wd = sd - (float)i0;
  int i1 = min(i0 + 1, F_ - 1);
  return (1.f - wd) * magbt[(size_t)i0 * NT] + wd * magbt[(size_t)i1 * NT];
}

__global__ __launch_bounds__(256) void k_harm(const float* __restrict__ mag,
                                              float* __restrict__ out) {
  long i = (long)blockIdx.x * blockDim.x + threadIdx.x;
  const long total = (long)B_ * F_ * NT;
  if (i >= total) return;
  int t = (int)(i % NT);
  long r = i / NT;
  int f = (int)(r % F_);
  int b = (int)(r / F_);
  const float* magbt = mag + (size_t)b * F_ * NT + t;   // index by [g*NT]

  float hs = log1pf(magbt[(size_t)f * NT]);
  #pragma unroll
  for (int k = 2; k <= 3; ++k) {
    int Fk = F_ / k;
    float su = ((float)f + 0.5f) * ((float)Fk / (float)F_) - 0.5f;
    su = fminf(fmaxf(su, 0.f), (float)(Fk - 1));
    int j0 = (int)su;
    float wu = su - (float)j0;
    int j1 = min(j0 + 1, Fk - 1);
    float v = (1.f - wu) * down_val(magbt, j0, Fk) + wu * down_val(magbt, j1, Fk);
    hs += log1pf(v);
  }
  out[(((size_t)b * 6 + 2) * F_ + f) * NT + t] = expm1f(hs);  // plane 2
}

// ---------------- launch ----------------
extern "C" void kernel_launch(void* const* d_in, const int* in_sizes, int n_in,
                              void* d_out, int out_size, void* d_ws, size_t ws_size,
                              hipStream_t stream) {
  const float* x = (const float*)d_in[0];
  float* out = (float*)d_out;
  float* ws = (float*)d_ws;

  size_t off = 0;
  float* bre    = ws + off; off += (size_t)FP * NFFT;
  float* bim    = ws + off; off += (size_t)FP * NFFT;
  float* spread = ws + off; off += (size_t)FP * FP;
  float* mag    = ws + off; off += (size_t)B_ * F_ * NT;
  float* mdb    = ws + off; off += (size_t)B_ * F_ * NT;

  k_fill_basis<<<(FP * NFFT + 255) / 256, 256, 0, stream>>>(bre, bim);
  k_fill_spread<<<FP, 256, 0, stream>>>(spread);

  dim3 gs(NTILE_N, B_);
  size_t lds_stft = (size_t)2 * NFFT * 17 * sizeof(float);   // 139,264 B (<320KB WGP)
  k_stft<<<gs, 256, lds_stft, stream>>>(x, bre, bim, out, mag, mdb);

  k_psy<<<gs, 256, 0, stream>>>(spread, mdb, out);

  long total = (long)B_ * F_ * NT;
  k_harm<<<(int)((total + 255) / 256), 256, 0, stream>>>(mag, out);
}